// GNN_51324859187767
// MI455X (gfx1250) — compile-verified
//
#include <hip/hip_runtime.h>

#define N_NODES 50000
#define HIDDEN  64
#define N_EDGES 1200000

typedef __attribute__((ext_vector_type(2))) float v2f;
typedef __attribute__((ext_vector_type(8))) float v8f;

// ---------------------------------------------------------------- zero fill
__global__ void GNN_zero_f4(float4* __restrict__ p, int n4) {
    int i = blockIdx.x * blockDim.x + threadIdx.x;
    if (i < n4) p[i] = make_float4(0.f, 0.f, 0.f, 0.f);
}

// ---------------------------------------------------------------- degree
__global__ void GNN_degree(const long long* __restrict__ dst, float* __restrict__ cnt) {
    int e = blockIdx.x * blockDim.x + threadIdx.x;
    if (e < N_EDGES) unsafeAtomicAdd(&cnt[(int)dst[e]], 1.0f);
}

// ---------------------------------------------------------------- scatter: agg[dst] += x[src]
// one thread per (edge, 4-float chunk); float4 gather + 4 hardware f32 atomics
__global__ void GNN_scatter(const float* __restrict__ X,
                            const long long* __restrict__ src,
                            const long long* __restrict__ dst,
                            float* __restrict__ AGG) {
    int t = blockIdx.x * blockDim.x + threadIdx.x;
    int e = t >> 4;
    if (e >= N_EDGES) return;
    int c = (t & 15) * 4;
    int s = (int)src[e];
    int d = (int)dst[e];
    float4 v = *(const float4*)(X + (size_t)s * HIDDEN + c);
    float* o = AGG + (size_t)d * HIDDEN + c;
    unsafeAtomicAdd(o + 0, v.x);
    unsafeAtomicAdd(o + 1, v.y);
    unsafeAtomicAdd(o + 2, v.z);
    unsafeAtomicAdd(o + 3, v.w);
}

// ---------------------------------------------------------------- fused GEMM
// OUT = relu( X @ Wself + (AGG * rowscale) @ Wnbr + bias )
// block = 128 thr = 4 waves; block -> 16-row tile, wave w -> 16-col tile w.
// K decomposed into 16 steps of V_WMMA_F32_16X16X4_F32.
__global__ __launch_bounds__(128)
void GNN_gemm_fused(const float* __restrict__ X, const float* __restrict__ AGG,
                    const float* __restrict__ Wself, const float* __restrict__ Wnbr,
                    const float* __restrict__ bias, const float* __restrict__ cnt,
                    float* __restrict__ OUT) {
    __shared__ float sWs[HIDDEN * HIDDEN];
    __shared__ float sWn[HIDDEN * HIDDEN];
    for (int i = threadIdx.x; i < (HIDDEN * HIDDEN / 4); i += 128) {
        ((float4*)sWs)[i] = ((const float4*)Wself)[i];
        ((float4*)sWn)[i] = ((const float4*)Wnbr)[i];
    }
    __syncthreads();

    const int lane = threadIdx.x & 31;
    const int wave = threadIdx.x >> 5;   // column tile index j = 0..3
    const int hi   = lane >> 4;          // half-wave: selects K pair / M+8
    const int lr   = lane & 15;
    const int rt   = blockIdx.x;         // row tile (3125 tiles cover 50000 rows)
    const int row  = rt * 16 + lr;       // A-matrix row held by this lane
    const int wcol = wave * 16 + lr;     // B/C column held by this lane

    // Preload B fragments (ISA 32-bit layout: lane=N; VGPR v holds K = 4k + v + 2*hi)
    v2f Bs[16], Bn[16];
#pragma unroll
    for (int k = 0; k < 16; ++k) {
        int k0 = 4 * k + 2 * hi;
        Bs[k].x = sWs[(k0    ) * HIDDEN + wcol];
        Bs[k].y = sWs[(k0 + 1) * HIDDEN + wcol];
        Bn[k].x = sWn[(k0    ) * HIDDEN + wcol];
        Bn[k].y = sWn[(k0 + 1) * HIDDEN + wcol];
    }

    float inv = 1.0f;
    if (cnt) inv = 1.0f / fmaxf(cnt[row], 1.0f);   // RGCN mean aggregation

    v8f acc;
    float bv = bias[wcol];
#pragma unroll
    for (int v = 0; v < 8; ++v) acc[v] = bv;

    const float* xr = X   + (size_t)row * HIDDEN;
    const float* gr = AGG + (size_t)row * HIDDEN;
#pragma unroll
    for (int k = 0; k < 16; ++k) {
        int k0 = 4 * k + 2 * hi;
        v2f a = *(const v2f*)(xr + k0);
        v2f g = *(const v2f*)(gr + k0);
        g *= inv;
        acc = __builtin_amdgcn_wmma_f32_16x16x4_f32(false, a, false, Bs[k],
                                                    (short)0, acc, false, false);
        acc = __builtin_amdgcn_wmma_f32_16x16x4_f32(false, g, false, Bn[k],
                                                    (short)0, acc, false, false);
    }

    // C/D layout: lane holds N=lr(+tile), VGPR v holds M = v + 8*hi
#pragma unroll
    for (int v = 0; v < 8; ++v) {
        float r = fmaxf(acc[v], 0.0f);
        OUT[(size_t)(rt * 16 + v + 8 * hi) * HIDDEN + wcol] = r;
    }
}

// ---------------------------------------------------------------- launcher
extern "C" void kernel_launch(void* const* d_in, const int* in_sizes, int n_in,
                              void* d_out, int out_size, void* d_ws, size_t ws_size,
                              hipStream_t stream) {
    const float*     x    = (const float*)d_in[0];
    const long long* ei   = (const long long*)d_in[1];
    const long long* src  = ei;
    const long long* dst  = ei + N_EDGES;
    const float* W_rel1  = (const float*)d_in[2];
    const float* W_root1 = (const float*)d_in[3];
    const float* b1      = (const float*)d_in[4];
    const float* W_self2 = (const float*)d_in[5];
    const float* W_nbr2  = (const float*)d_in[6];
    const float* b2      = (const float*)d_in[7];
    const float* W_self3 = (const float*)d_in[8];
    const float* W_nbr3  = (const float*)d_in[9];
    const float* b3      = (const float*)d_in[10];
    const float* W_self4 = (const float*)d_in[11];
    const float* W_nbr4  = (const float*)d_in[12];
    const float* b4      = (const float*)d_in[13];

    float* out1 = (float*)d_out;               // x_1 (after layer 2)
    float* out2 = out1 + (size_t)N_NODES * HIDDEN;  // x_2 (after layer 4)

    float* agg = (float*)d_ws;                             // N*H
    float* h   = agg + (size_t)N_NODES * HIDDEN;           // N*H
    float* cnt = h   + (size_t)N_NODES * HIDDEN;           // N

    const int NH4   = N_NODES * HIDDEN / 4;                 // 800000
    const int ZB    = (NH4 + 255) / 256;
    const int ZC    = (N_NODES / 4 + 255) / 256;
    const int EB    = (N_EDGES + 255) / 256;
    const int SB    = (N_EDGES * 16 + 255) / 256;
    const int GB    = N_NODES / 16;                         // 3125 row tiles

    // degree (once)
    GNN_zero_f4<<<ZC, 256, 0, stream>>>((float4*)cnt, N_NODES / 4);
    GNN_degree<<<EB, 256, 0, stream>>>(dst, cnt);

    // Layer 1 (RGCN): h = relu(x @ W_root1 + mean_agg(x) @ W_rel1 + b1)
    GNN_zero_f4<<<ZB, 256, 0, stream>>>((float4*)agg, NH4);
    GNN_scatter<<<SB, 256, 0, stream>>>(x, src, dst, agg);
    GNN_gemm_fused<<<GB, 128, 0, stream>>>(x, agg, W_root1, W_rel1, b1, cnt, h);

    // Layer 2 (GraphConv): x_1 = relu(h @ W_self2 + sum_agg(h) @ W_nbr2 + b2)
    GNN_zero_f4<<<ZB, 256, 0, stream>>>((float4*)agg, NH4);
    GNN_scatter<<<SB, 256, 0, stream>>>(h, src, dst, agg);
    GNN_gemm_fused<<<GB, 128, 0, stream>>>(h, agg, W_self2, W_nbr2, b2, nullptr, out1);

    // Layer 3: h = relu(x_1 @ W_self3 + sum_agg(x_1) @ W_nbr3 + b3)
    GNN_zero_f4<<<ZB, 256, 0, stream>>>((float4*)agg, NH4);
    GNN_scatter<<<SB, 256, 0, stream>>>(out1, src, dst, agg);
    GNN_gemm_fused<<<GB, 128, 0, stream>>>(out1, agg, W_self3, W_nbr3, b3, nullptr, h);

    // Layer 4: x_2 = relu(h @ W_self4 + sum_agg(h) @ W_nbr4 + b4)
    GNN_zero_f4<<<ZB, 256, 0, stream>>>((float4*)agg, NH4);
    GNN_scatter<<<SB, 256, 0, stream>>>(h, src, dst, agg);
    GNN_gemm_fused<<<GB, 128, 0, stream>>>(h, agg, W_self4, W_nbr4, b4, nullptr, out2);
}